// GNN_62981400429144
// MI455X (gfx1250) — compile-verified
//
#include <hip/hip_runtime.h>

typedef __attribute__((ext_vector_type(2))) float v2f;
typedef __attribute__((ext_vector_type(8))) float v8f;

#define N_NODES 100000
#define N_EDGES 1600000
#define D       128
#define LDSS    132   // LDS row stride (pad 128 -> 132 to spread 64 banks)

// ---------------- utility kernels ----------------

__global__ void zero_f4(float4* __restrict__ p, int n4) {
    int i = blockIdx.x * blockDim.x + threadIdx.x;
    if (i < n4) p[i] = make_float4(0.f, 0.f, 0.f, 0.f);
}

__global__ void deg_count(const int* __restrict__ dst, float* __restrict__ deg) {
    int e = blockIdx.x * blockDim.x + threadIdx.x;
    if (e < N_EDGES) atomicAdd(&deg[dst[e]], 1.0f);
}

__global__ void deg_invert(float* __restrict__ deg) {
    int i = blockIdx.x * blockDim.x + threadIdx.x;
    if (i < N_NODES) {
        float d = deg[i];
        deg[i] = (d > 0.f) ? (1.0f / d) : 0.0f;
    }
}

// Gather x[src] rows, scatter-add into msg[dst]. One thread = 4 features of 1 edge.
__global__ void scatter_add(const float* __restrict__ feat, const int* __restrict__ src,
                            const int* __restrict__ dst, float* __restrict__ msg) {
    int tid = blockIdx.x * blockDim.x + threadIdx.x;   // N_EDGES*32 threads total
    int e = tid >> 5;
    if (e >= N_EDGES) return;
    int f = (tid & 31) << 2;
    int s = src[e], d = dst[e];
    const float4 v = *reinterpret_cast<const float4*>(feat + (size_t)s * D + f);
    float* p = msg + (size_t)d * D + f;
    atomicAdd(p + 0, v.x);
    atomicAdd(p + 1, v.y);
    atomicAdd(p + 2, v.z);
    atomicAdd(p + 3, v.w);
}

__global__ void copy_i32(const int* __restrict__ s, int* __restrict__ d, int n) {
    int i = blockIdx.x * blockDim.x + threadIdx.x;
    if (i < n) d[i] = s[i];
}

// ---------------- fused SAGE layer GEMM ----------------
// out[16-row strip] = relu?( (msg*deg_inv) @ Wl + bias + xin @ Wr )
// Block: 256 threads (8 wave32). Wave w computes the 16x16 tile at columns [16w,16w+16).
// Uses V_WMMA_F32_16X16X4_F32, K-loop of 32 steps, 2 WMMA per step chained into one acc.
__global__ void __launch_bounds__(256)
sage_gemm(const float* __restrict__ msg, const float* __restrict__ dinv,
          const float* __restrict__ xin, const float* __restrict__ Wl,
          const float* __restrict__ bias, const float* __restrict__ Wr,
          float* __restrict__ out, int do_relu) {
    __shared__ float sAgg[16 * LDSS];
    __shared__ float sX[16 * LDSS];

    const int tile = blockIdx.x;                 // 0..6249 (100000/16)
    const size_t base = (size_t)tile * 16 * D;

    // Stage 16x128 strips of agg (scaled by deg_inv) and x into LDS.
    for (int i = threadIdx.x; i < 16 * D; i += 256) {
        int r = i >> 7;          // row within strip
        int c = i & (D - 1);     // column
        float scale = dinv[tile * 16 + r];
        sAgg[r * LDSS + c] = msg[base + i] * scale;
        sX[r * LDSS + c]   = xin[base + i];
    }
    __syncthreads();

    const int wave = threadIdx.x >> 5;           // N-tile index, 0..7
    const int lane = threadIdx.x & 31;
    const int l    = lane & 15;
    const int koff = (lane >> 4) << 1;           // lanes 16-31 carry K+2,K+3
    const int n    = wave * 16 + l;              // output column

    v8f acc = {};
    for (int k = 0; k < D; k += 4) {
        const int kk = k + koff;
        // A fragments (16x4 f32): lane l holds row M=l, elems K=kk, kk+1
        v2f aA = { sAgg[l * LDSS + kk], sAgg[l * LDSS + kk + 1] };
        v2f aX = { sX[l * LDSS + kk],   sX[l * LDSS + kk + 1] };
        // B fragments (4x16 f32): VGPR0 = row K=kk, VGPR1 = row K=kk+1, col n
        v2f bL = { Wl[(size_t)kk * D + n], Wl[(size_t)(kk + 1) * D + n] };
        v2f bR = { Wr[(size_t)kk * D + n], Wr[(size_t)(kk + 1) * D + n] };
        acc = __builtin_amdgcn_wmma_f32_16x16x4_f32(
                  false, aA, false, bL, (short)0, acc, false, false);
        acc = __builtin_amdgcn_wmma_f32_16x16x4_f32(
                  false, aX, false, bR, (short)0, acc, false, false);
    }

    // Epilogue: bias + optional relu + store per documented C/D layout:
    // VGPR v, lanes 0-15 -> M=v, lanes 16-31 -> M=v+8; N = lane&15 (+16w).
    const float bv = bias[n];
    const int rbase = tile * 16 + ((lane >> 4) << 3);
#pragma unroll
    for (int v = 0; v < 8; ++v) {
        float o = acc[v] + bv;
        if (do_relu) o = fmaxf(o, 0.0f);
        out[(size_t)(rbase + v) * D + n] = o;
    }
}

// ---------------- launcher ----------------

extern "C" void kernel_launch(void* const* d_in, const int* in_sizes, int n_in,
                              void* d_out, int out_size, void* d_ws, size_t ws_size,
                              hipStream_t stream) {
    const float* x   = (const float*)d_in[0];
    const float* W1l = (const float*)d_in[1];
    const float* b1  = (const float*)d_in[2];
    const float* W1r = (const float*)d_in[3];
    const float* W2l = (const float*)d_in[4];
    const float* b2  = (const float*)d_in[5];
    const float* W2r = (const float*)d_in[6];
    const int* eidx  = (const int*)d_in[7];
    const int* src   = eidx;              // edge_index[0]
    const int* dst   = eidx + N_EDGES;    // edge_index[1]

    float* outp = (float*)d_out;

    // workspace layout: deg(100000) | msg(100000*128) | h(100000*128)  ~= 103 MB
    float* deg  = (float*)d_ws;
    float* msgb = deg + N_NODES;
    float* h    = msgb + (size_t)N_NODES * D;

    const int feat4 = N_NODES * D / 4;                // 3,200,000 float4
    const int scatterBlocks = (N_EDGES * 32) / 256;   // 200,000

    // degrees -> deg_inv (in place)
    zero_f4<<<(N_NODES / 4 + 255) / 256, 256, 0, stream>>>((float4*)deg, N_NODES / 4);
    deg_count<<<(N_EDGES + 255) / 256, 256, 0, stream>>>(dst, deg);
    deg_invert<<<(N_NODES + 255) / 256, 256, 0, stream>>>(deg);

    // ---- layer 1 ----
    zero_f4<<<(feat4 + 255) / 256, 256, 0, stream>>>((float4*)msgb, feat4);
    scatter_add<<<scatterBlocks, 256, 0, stream>>>(x, src, dst, msgb);
    sage_gemm<<<N_NODES / 16, 256, 0, stream>>>(msgb, deg, x, W1l, b1, W1r, h, 1);

    // ---- layer 2 ----
    zero_f4<<<(feat4 + 255) / 256, 256, 0, stream>>>((float4*)msgb, feat4);
    scatter_add<<<scatterBlocks, 256, 0, stream>>>(h, src, dst, msgb);
    sage_gemm<<<N_NODES / 16, 256, 0, stream>>>(msgb, deg, h, W2l, b2, W2r, outp, 0);

    // tuple output: append edge_index (bit-copy int32 into float-typed buffer tail)
    copy_i32<<<(2 * N_EDGES + 255) / 256, 256, 0, stream>>>(
        eidx, (int*)d_out + (size_t)N_NODES * D, 2 * N_EDGES);
}